// SURE_pure4D_78426102825224
// MI455X (gfx1250) — compile-verified
//
#include <hip/hip_runtime.h>
#include <math.h>

// ---------------- problem constants ----------------
#define EPSF     1e-6f
#define NB       32
#define NCLS     16
#define NDIST    8
#define PIX      16384          // C*H*W = 64*16*16
#define KPB      32768          // 2*PIX (tm or xy block per batch)
#define BSTRIDE  65536          // 4*PIX per batch in x_LE

// ---------------- workspace layout (float offsets), total 1,704,216 f ----
#define XLE_OFF  0u             // xle_out   [16*32768]
#define XY_OFF   524288u        // xy_out    [16*32768]
#define MT_OFF   1048576u       // means_theta [16*16384]
#define LMM_OFF  1310720u       // log(means_mag+EPS) [16*16384]
#define XW_OFF   1572864u       // X_w [16]
#define W1N_OFF  1572880u       // w1n [8]
#define A_OFF    1572888u       // a[c][d] [128]
#define BB_OFF   1573016u       // b[c][d] [128]
#define LEP_OFF  1573144u       // le partials [1024 blocks][128]

typedef float v2f __attribute__((ext_vector_type(2)));
typedef float v8f __attribute__((ext_vector_type(8)));

__device__ __forceinline__ float lsig(float x) {
    // numerically stable log_sigmoid
    return fminf(x, 0.0f) - log1pf(expf(-fabsf(x)));
}

// ---------------- K0: tiny tables (1 block) ----------------
__global__ void k0_prep(const int* __restrict__ labels,
                        const float* __restrict__ X_weights,
                        const float* __restrict__ sigmas,
                        const float* __restrict__ w1,
                        const float* __restrict__ tao,
                        float* __restrict__ ws) {
    const int t = threadIdx.x;
    if (t < NDIST) {
        float s = 0.f;
        for (int d = 0; d < NDIST; ++d) s += w1[d] * w1[d];
        ws[W1N_OFF + t] = (w1[t] * w1[t]) / s;
    }
    if (t < NCLS) {
        int cnt = 0;
        for (int b = 0; b < NB; ++b) cnt += (labels[b] == t) ? 1 : 0;
        ws[XW_OFF + t] = X_weights[t] + (float)cnt;
    }
    if (t < NCLS * NDIST) {
        const int c = t >> 3, d = t & 7;
        const float t2 = tao[d] * tao[d];
        const float s2 = sigmas[c] * sigmas[c];
        const float inv = 1.0f / (s2 + t2);
        ws[A_OFF + t]  = t2 * inv;   // a[c,d]
        ws[BB_OFF + t] = s2 * inv;   // b[c,d]
    }
}

// ---------------- K1: segment-sum as 16x32 x 32xN GEMM via f32 WMMA ------
// One wave computes one 16(class) x 16(column) tile of S = onehot * X,
// accumulating K=32 (batch) in 8 V_WMMA_F32_16X16X4_F32 steps, then
// normalizes with X_w and writes xle_out / xy_out.
__global__ void k1_segsum_wmma(const float* __restrict__ x,
                               const int* __restrict__ labels,
                               const float* __restrict__ X_LEs,
                               const float* __restrict__ X_LEs_xy,
                               float* __restrict__ ws) {
    const int lane  = threadIdx.x & 31;
    const int wave  = threadIdx.x >> 5;
    const int tile  = blockIdx.x * 4 + wave;     // 0..4095
    const int hi    = lane >> 4;                 // half-wave select
    const int col   = lane & 15;

    const bool is_xy  = (tile >= 2048);
    const int  k0     = (tile & 2047) * 16;      // column base within 32768
    const int  srcoff = is_xy ? KPB : 0;         // tm = ch 0..1, xy = ch 2..3

    v8f acc = {};
#pragma unroll
    for (int s = 0; s < 8; ++s) {
        // A fragment (16x4, M=class, K=batch): VGPR0 = K {4s+0 | 4s+2},
        // VGPR1 = K {4s+1 | 4s+3} for lane halves.
        const int b0 = 4 * s + 2 * hi;
        v2f a, bf;
        a.x = (labels[b0]     == col) ? 1.0f : 0.0f;
        a.y = (labels[b0 + 1] == col) ? 1.0f : 0.0f;
        // B fragment (4x16, K=batch rows striped across lanes per VGPR)
        const float* xc = x + (size_t)b0 * BSTRIDE + srcoff + (k0 + col);
        bf.x = xc[0];
        bf.y = xc[BSTRIDE];
        acc = __builtin_amdgcn_wmma_f32_16x16x4_f32(
            false, a, false, bf, (short)0, acc, false, false);
    }

    const float* base = is_xy ? X_LEs_xy : X_LEs;
    float* dst = ws + (is_xy ? XY_OFF : XLE_OFF);
#pragma unroll
    for (int r = 0; r < 8; ++r) {
        const int cls = r + 8 * hi;              // D: VGPR r = rows r / r+8
        const int idx = cls * KPB + k0 + col;
        dst[idx] = (base[idx] + acc[r]) / ws[XW_OFF + cls];
    }
}

// ---------------- K2: means_theta, log(means_mag+eps), le partials -------
// Block = 16 classes x 16 pixels; miu-derived log-sigmoids shared via LDS.
__global__ void k2_means(const float* __restrict__ miu,
                         float* __restrict__ ws) {
    __shared__ float s_m0 [NDIST][16];  // miu[d,0,p]
    __shared__ float s_l0 [NDIST][16];  // ls(miu[d,0,p])
    __shared__ float s_l1 [NDIST][16];  // ls(miu[d,1,p])
    __shared__ float s_l1e[NDIST][16];  // ls(miu[d,1,p]+eps)
    __shared__ float s_le [NCLS][NDIST][16];

    const int tid = threadIdx.x;                 // 256
    const int p0  = blockIdx.x * 16;
    if (tid < NDIST * 16) {
        const int d = tid >> 4, pj = tid & 15;
        const float m0 = miu[d * KPB + p0 + pj];
        const float m1 = miu[d * KPB + PIX + p0 + pj];
        s_m0 [d][pj] = m0;
        s_l0 [d][pj] = lsig(m0);
        s_l1 [d][pj] = lsig(m1);
        s_l1e[d][pj] = lsig(m1 + EPSF);
    }
    __syncthreads();

    const int c = tid >> 4, pi = tid & 15, p = p0 + pi;
    const float x0 = ws[XLE_OFF + c * KPB + p];
    const float x1 = ws[XLE_OFF + c * KPB + PIX + p];   // mag
    const float lsx0 = lsig(x0);
    const float lsx1 = lsig(x1);
    const float lsE  = lsig(x1 + EPSF);

    float mt = 0.f, mm = 0.f;
#pragma unroll
    for (int d = 0; d < NDIST; ++d) {
        const float av = ws[A_OFF  + c * 8 + d];
        const float bv = ws[BB_OFF + c * 8 + d];
        const float wn = ws[W1N_OFF + d];
        mt += (x1 * av + s_m0[d][pi] * bv) * wn;
        mm += expf((av * lsE + bv * s_l1e[d][pi]) * wn);
        const float d0 = lsx0 - s_l0[d][pi];
        const float d1 = lsx1 - s_l1[d][pi];
        s_le[c][d][pi] = d0 * d0 + d1 * d1;
    }
    ws[MT_OFF  + c * PIX + p] = mt;
    ws[LMM_OFF + c * PIX + p] = logf(mm + EPSF);
    __syncthreads();

    // deterministic per-block partial for le_norm[c][d]
    if (tid < NCLS * NDIST) {
        const int cc = tid >> 3, dd = tid & 7;
        float s = 0.f;
#pragma unroll
        for (int pj = 0; pj < 16; ++pj) s += s_le[cc][dd][pj];
        ws[LEP_OFF + (size_t)blockIdx.x * 128 + tid] = s;
    }
}

// ---------------- K3: distance + min over classes ------------------------
// Block = 32 batches x 8 pixels; per-class values cached in LDS.
__global__ void k3_dist(const float* __restrict__ x,
                        const float* __restrict__ weight,
                        const float* __restrict__ ws,
                        float* __restrict__ out) {
    __shared__ float s_mt [NCLS][8];
    __shared__ float s_lmm[NCLS][8];
    __shared__ float s_xy0[NCLS][8];
    __shared__ float s_xy1[NCLS][8];

    const int tid = threadIdx.x;                 // 256
    const int p0  = blockIdx.x * 8;
    for (int i = tid; i < 512; i += 256) {
        const int arr = i >> 7, rest = i & 127;
        const int c = rest >> 3, pj = rest & 7;
        const int p = p0 + pj;
        if      (arr == 0) s_mt [c][pj] = ws[MT_OFF  + c * PIX + p];
        else if (arr == 1) s_lmm[c][pj] = ws[LMM_OFF + c * PIX + p];
        else if (arr == 2) s_xy0[c][pj] = ws[XY_OFF + c * KPB + p];
        else               s_xy1[c][pj] = ws[XY_OFF + c * KPB + PIX + p];
    }
    __syncthreads();

    const int b = tid & 31, pi = tid >> 5;
    const int p = p0 + pi;
    const float* xb = x + (size_t)b * BSTRIDE;
    const float xt0 = xb[p];
    const float xt1 = xb[PIX + p];
    const float xx0 = xb[2 * PIX + p];
    const float xx1 = xb[3 * PIX + p];
    const float lx  = logf(xt1);
    const float w0  = weight[0] * weight[0];
    const float w1w = weight[1] * weight[1];
    const float w2  = weight[2] * weight[2];

    float best = 3.402823466e38f;
#pragma unroll
    for (int c = 0; c < NCLS; ++c) {
        const float dr  = fabsf(xt0 - s_mt[c][pi]);
        const float da  = fabsf(lx - s_lmm[c][pi]);
        const float dx0 = xx0 - s_xy0[c][pi];
        const float dx1 = xx1 - s_xy1[c][pi];
        const float dv  = w0 * dr + w1w * da + w2 * (dx0 * dx0 + dx1 * dx1);
        best = fminf(best, dv);
    }
    out[(size_t)b * PIX + p] = best;
}

// ---------------- K4: loss (deterministic reduction of le partials) ------
__global__ void k4_loss(const float* __restrict__ sigmas,
                        const float* __restrict__ tao,
                        const float* __restrict__ ws,
                        float* __restrict__ loss_out) {
    __shared__ float red[NCLS][NDIST];
    const int t = threadIdx.x;                   // 128: (c,d)
    const int c = t >> 3, d = t & 7;
    float le = 0.f;
    for (int blk = 0; blk < 1024; ++blk)
        le += ws[LEP_OFF + (size_t)blk * 128 + t];

    const float s2 = sigmas[c] * sigmas[c];
    const float t2 = tao[d] * tao[d];
    const float term1 = s2 / ((t2 + s2) * (t2 + s2));
    const float term2 = s2 * le;
    const float term3 = (2.0f * (float)PIX) * (t2 * t2 - s2 * s2) / ws[XW_OFF + c];
    red[c][d] = term1 * (term2 + term3);
    __syncthreads();
    if (t < NDIST) {
        float s = 0.f;
        for (int cc = 0; cc < NCLS; ++cc) s += red[cc][t];
        loss_out[t] = s / (float)NCLS;
    }
}

// ---------------- host launch -------------------------------------------
extern "C" void kernel_launch(void* const* d_in, const int* in_sizes, int n_in,
                              void* d_out, int out_size, void* d_ws, size_t ws_size,
                              hipStream_t stream) {
    const float* x_LE      = (const float*)d_in[0];
    const int*   labels    = (const int*)  d_in[1];
    const float* X_LEs     = (const float*)d_in[2];
    const float* X_LEs_xy  = (const float*)d_in[3];
    const float* X_weights = (const float*)d_in[4];
    const float* sigmas    = (const float*)d_in[5];
    const float* w1        = (const float*)d_in[6];
    const float* miu       = (const float*)d_in[7];
    const float* tao       = (const float*)d_in[8];
    const float* weight    = (const float*)d_in[9];
    float* out = (float*)d_out;          // [32*16384] out, then [8] loss
    float* ws  = (float*)d_ws;

    k0_prep      <<<1,    128, 0, stream>>>(labels, X_weights, sigmas, w1, tao, ws);
    k1_segsum_wmma<<<1024, 128, 0, stream>>>(x_LE, labels, X_LEs, X_LEs_xy, ws);
    k2_means     <<<1024, 256, 0, stream>>>(miu, ws);
    k3_dist      <<<2048, 256, 0, stream>>>(x_LE, weight, ws, out);
    k4_loss      <<<1,    128, 0, stream>>>(sigmas, tao, ws, out + (size_t)NB * PIX);
}